// GnnAgent_29214367547987
// MI455X (gfx1250) — compile-verified
//
#include <hip/hip_runtime.h>

#define N_SRC   50000
#define N_DST   50000
#define E_EDGES 800000
#define D_IN    64
#define HID     256
#define HEADS   4
#define N_ACT   16
#define NEG_SLOPE 0.2f

typedef float v2f __attribute__((ext_vector_type(2)));
typedef float v8f __attribute__((ext_vector_type(8)));

__device__ __forceinline__ v8f wmma_f32_k4(v2f a, v2f b, v8f c) {
    // D = A(16x4 f32) * B(4x16 f32) + C(16x16 f32)
    return __builtin_amdgcn_wmma_f32_16x16x4_f32(
        /*neg_a=*/false, a, /*neg_b=*/false, b,
        /*c_mod=*/(short)0, c, /*reuse_a=*/false, /*reuse_b=*/false);
}

__device__ __forceinline__ float sigmoidf_(float x) {
    return 1.f / (1.f + expf(-x));
}

// ---------------------------------------------------------------------------
// C[M,256] = A[M,64] @ W[64,256] + bias ; 32-row tile per block (two 16-row
// WMMA tiles sharing each B fragment), 8 waves x two 16-col tiles.
// ---------------------------------------------------------------------------
__global__ __launch_bounds__(256)
void proj_gemm(const float* __restrict__ A, const float* __restrict__ W,
               const float* __restrict__ bias, float* __restrict__ C, int M) {
    __shared__ float as[32 * D_IN];
    const int tid = threadIdx.x;
    const int m0  = blockIdx.x * 32;

    for (int idx = tid; idx < 32 * D_IN; idx += 256) {
        const int row = m0 + (idx >> 6);
        as[idx] = (row < M) ? A[(long)row * D_IN + (idx & 63)] : 0.f;
    }
    __syncthreads();

    const int wave = tid >> 5, lane = tid & 31;
    const int half = lane >> 4, lid = lane & 15;
    const bool tile1 = (m0 + 16 < M);   // block-uniform

    for (int t = 0; t < 2; ++t) {
        const int n0 = (wave + t * 8) * 16;
        v8f acc0 = {}, acc1 = {};
        #pragma unroll
        for (int kk = 0; kk < D_IN; kk += 4) {
            const int k = kk + 2 * half;
            const v2f a0 = *(const v2f*)&as[lid * D_IN + k];
            const v2f a1 = *(const v2f*)&as[(16 + lid) * D_IN + k];
            v2f b;
            b.x = W[k * HID + n0 + lid];
            b.y = W[(k + 1) * HID + n0 + lid];
            acc0 = wmma_f32_k4(a0, b, acc0);
            acc1 = wmma_f32_k4(a1, b, acc1);
        }
        const float bv = bias[n0 + lid];
        #pragma unroll
        for (int r = 0; r < 8; ++r)
            C[(long)(m0 + 8 * half + r) * HID + n0 + lid] = acc0[r] + bv;
        if (tile1) {
            #pragma unroll
            for (int r = 0; r < 8; ++r)
                C[(long)(m0 + 16 + 8 * half + r) * HID + n0 + lid] = acc1[r] + bv;
        }
    }
}

// ---------------------------------------------------------------------------
// Init: zero rst (d_out h-region) and den, smax = -inf (ordered-int encoding)
// ---------------------------------------------------------------------------
__global__ __launch_bounds__(256)
void init_kernel(float* __restrict__ rst, unsigned* __restrict__ smax,
                 float* __restrict__ den) {
    const int idx = blockIdx.x * 256 + threadIdx.x;
    if (idx < N_DST * HID) rst[idx] = 0.f;
    if (idx < N_DST * HEADS) { smax[idx] = 0xFF800000u; den[idx] = 0.f; }
}

__device__ __forceinline__ void atomicMaxF(unsigned* addr, float v) {
    const int iv = __float_as_int(v);
    if (iv >= 0) atomicMax((int*)addr, iv);
    else         atomicMin(addr, (unsigned)iv);
}

// ---------------------------------------------------------------------------
// Per-edge scores: one wave per edge; lane l handles feats [8l, 8l+8);
// head = l/8; 8-lane shuffle reduction; atomic segment-max by dst.
// ---------------------------------------------------------------------------
__global__ __launch_bounds__(256)
void edge_score(const float* __restrict__ fs, const float* __restrict__ fd,
                const float* __restrict__ attn, const int* __restrict__ esrc,
                const int* __restrict__ edst, float* __restrict__ score,
                unsigned* __restrict__ smax) {
    const int e    = blockIdx.x * 8 + (threadIdx.x >> 5);
    const int lane = threadIdx.x & 31;
    if (e >= E_EDGES) return;
    const int s = esrc[e], d = edst[e];
    const float4* pfs = (const float4*)(fs + (long)s * HID);
    const float4* pfd = (const float4*)(fd + (long)d * HID);
    const float4* pat = (const float4*)attn;
    float acc = 0.f;
    #pragma unroll
    for (int q = 0; q < 2; ++q) {
        const float4 a = pfs[lane * 2 + q];
        const float4 b = pfd[lane * 2 + q];
        const float4 w = pat[lane * 2 + q];
        float e0 = a.x + b.x; e0 = e0 > 0.f ? e0 : NEG_SLOPE * e0; acc += e0 * w.x;
        float e1 = a.y + b.y; e1 = e1 > 0.f ? e1 : NEG_SLOPE * e1; acc += e1 * w.y;
        float e2 = a.z + b.z; e2 = e2 > 0.f ? e2 : NEG_SLOPE * e2; acc += e2 * w.z;
        float e3 = a.w + b.w; e3 = e3 > 0.f ? e3 : NEG_SLOPE * e3; acc += e3 * w.w;
    }
    acc += __shfl_xor(acc, 1, 32);
    acc += __shfl_xor(acc, 2, 32);
    acc += __shfl_xor(acc, 4, 32);
    if ((lane & 7) == 0) {
        const int h = lane >> 3;
        score[(long)e * HEADS + h] = acc;
        atomicMaxF(&smax[d * HEADS + h], acc);
    }
}

// ---------------------------------------------------------------------------
// ex = exp(score - smax[dst]); score overwritten in-place; den += ex
// ---------------------------------------------------------------------------
__global__ __launch_bounds__(256)
void edge_exp(float* __restrict__ score, const unsigned* __restrict__ smax,
              float* __restrict__ den, const int* __restrict__ edst) {
    const long idx = (long)blockIdx.x * 256 + threadIdx.x;
    if (idx >= (long)E_EDGES * HEADS) return;
    const int e = (int)(idx >> 2), h = (int)(idx & 3);
    const int d = edst[e];
    const float mx = __int_as_float((int)smax[d * HEADS + h]);
    const float ex = expf(score[idx] - mx);
    score[idx] = ex;
    atomicAdd(&den[d * HEADS + h], ex);
}

// ---------------------------------------------------------------------------
// rst[dst] += alpha * fs[src]; one wave per edge, 8 floats per lane.
// ---------------------------------------------------------------------------
__global__ __launch_bounds__(256)
void edge_msg(const float* __restrict__ fs, const float* __restrict__ ex,
              const float* __restrict__ den, const int* __restrict__ esrc,
              const int* __restrict__ edst, float* __restrict__ rst) {
    const int e    = blockIdx.x * 8 + (threadIdx.x >> 5);
    const int lane = threadIdx.x & 31;
    if (e >= E_EDGES) return;
    const int s = esrc[e], d = edst[e];
    const int h = lane >> 3;
    const float alpha = ex[(long)e * HEADS + h] / den[d * HEADS + h];
    const float* src = fs + (long)s * HID + lane * 8;
    float* dst       = rst + (long)d * HID + lane * 8;
    #pragma unroll
    for (int q = 0; q < 8; ++q)
        atomicAdd(&dst[q], alpha * src[q]);
}

// ---------------------------------------------------------------------------
// Fused GRU over a 32-row tile: x = relu(rst + resid); gi = x@W_ih^T,
// gh = h0@W_hh^T via f32 WMMA. Each wave: 12 accumulators per 16-col chunk
// (6 gates x 2 row tiles) so every B fragment feeds two WMMAs.
// hout holds rst on entry, h_new on exit (block-local rows only).
// ---------------------------------------------------------------------------
__global__ __launch_bounds__(256)
void gru_kernel(const float* __restrict__ resid, const float* __restrict__ h0,
                const float* __restrict__ Wih, const float* __restrict__ Whh,
                const float* __restrict__ bih, const float* __restrict__ bhh,
                float* __restrict__ hout) {
    __shared__ float xs[32 * HID];
    __shared__ float hs[32 * HID];
    const int tid = threadIdx.x;
    const int m0  = blockIdx.x * 32;

    for (int idx = tid; idx < 32 * HID; idx += 256) {
        const int row = m0 + (idx >> 8);
        if (row < N_DST) {
            const long g = (long)row * HID + (idx & 255);
            const float x = hout[g] + resid[g];
            xs[idx] = x > 0.f ? x : 0.f;
            hs[idx] = h0[g];
        } else {
            xs[idx] = 0.f;
            hs[idx] = 0.f;
        }
    }
    __syncthreads();

    const int wave = tid >> 5, lane = tid & 31;
    const int half = lane >> 4, lid = lane & 15;
    const bool tile1 = (m0 + 16 < N_DST);   // block-uniform

    for (int t = 0; t < 2; ++t) {
        const int c = wave + t * 8;     // column chunk 0..15
        const int j = c * 16 + lid;     // column within [0,256)
        const float* wir = Wih + (long)j * HID;
        const float* wiz = Wih + (long)(HID + j) * HID;
        const float* win = Wih + (long)(2 * HID + j) * HID;
        const float* whr = Whh + (long)j * HID;
        const float* whz = Whh + (long)(HID + j) * HID;
        const float* whn = Whh + (long)(2 * HID + j) * HID;
        __builtin_prefetch(wir, 0, 1);
        __builtin_prefetch(wiz, 0, 1);
        __builtin_prefetch(win, 0, 1);
        __builtin_prefetch(whr, 0, 1);
        __builtin_prefetch(whz, 0, 1);
        __builtin_prefetch(whn, 0, 1);

        v8f air0 = {}, aiz0 = {}, ain0 = {}, ahr0 = {}, ahz0 = {}, ahn0 = {};
        v8f air1 = {}, aiz1 = {}, ain1 = {}, ahr1 = {}, ahz1 = {}, ahn1 = {};
        for (int kk = 0; kk < HID; kk += 4) {
            const int k = kk + 2 * half;
            const v2f ax0 = *(const v2f*)&xs[lid * HID + k];
            const v2f ah0 = *(const v2f*)&hs[lid * HID + k];
            const v2f ax1 = *(const v2f*)&xs[(16 + lid) * HID + k];
            const v2f ah1 = *(const v2f*)&hs[(16 + lid) * HID + k];
            v2f b;
            b = *(const v2f*)&wir[k];
            air0 = wmma_f32_k4(ax0, b, air0);
            air1 = wmma_f32_k4(ax1, b, air1);
            b = *(const v2f*)&wiz[k];
            aiz0 = wmma_f32_k4(ax0, b, aiz0);
            aiz1 = wmma_f32_k4(ax1, b, aiz1);
            b = *(const v2f*)&win[k];
            ain0 = wmma_f32_k4(ax0, b, ain0);
            ain1 = wmma_f32_k4(ax1, b, ain1);
            b = *(const v2f*)&whr[k];
            ahr0 = wmma_f32_k4(ah0, b, ahr0);
            ahr1 = wmma_f32_k4(ah1, b, ahr1);
            b = *(const v2f*)&whz[k];
            ahz0 = wmma_f32_k4(ah0, b, ahz0);
            ahz1 = wmma_f32_k4(ah1, b, ahz1);
            b = *(const v2f*)&whn[k];
            ahn0 = wmma_f32_k4(ah0, b, ahn0);
            ahn1 = wmma_f32_k4(ah1, b, ahn1);
        }
        const float bir = bih[j], biz = bih[HID + j], bin = bih[2 * HID + j];
        const float bhr = bhh[j], bhz = bhh[HID + j], bhn = bhh[2 * HID + j];
        #pragma unroll
        for (int r = 0; r < 8; ++r) {
            const int ml = 8 * half + r;
            const float rg = sigmoidf_((air0[r] + bir) + (ahr0[r] + bhr));
            const float zg = sigmoidf_((aiz0[r] + biz) + (ahz0[r] + bhz));
            const float ng = tanhf((ain0[r] + bin) + rg * (ahn0[r] + bhn));
            const float hold = hs[ml * HID + j];
            hout[(long)(m0 + ml) * HID + j] = (1.f - zg) * ng + zg * hold;
        }
        if (tile1) {
            #pragma unroll
            for (int r = 0; r < 8; ++r) {
                const int ml = 16 + 8 * half + r;
                const float rg = sigmoidf_((air1[r] + bir) + (ahr1[r] + bhr));
                const float zg = sigmoidf_((aiz1[r] + biz) + (ahz1[r] + bhz));
                const float ng = tanhf((ain1[r] + bin) + rg * (ahn1[r] + bhn));
                const float hold = hs[ml * HID + j];
                hout[(long)(m0 + ml) * HID + j] = (1.f - zg) * ng + zg * hold;
            }
        }
    }
}

// ---------------------------------------------------------------------------
// logits[16 nodes x 16 acts] per wave: K=256 f32 WMMA chain.
// ---------------------------------------------------------------------------
__global__ __launch_bounds__(256)
void logits_kernel(const float* __restrict__ h, const float* __restrict__ Wout,
                   const float* __restrict__ bout, float* __restrict__ out) {
    const int tile = blockIdx.x * 8 + (threadIdx.x >> 5);
    const int lane = threadIdx.x & 31;
    const int half = lane >> 4, lid = lane & 15;
    if (tile >= N_DST / 16) return;
    const int m0 = tile * 16;
    v8f acc = {};
    for (int kk = 0; kk < HID; kk += 4) {
        const int k = kk + 2 * half;
        const v2f a = *(const v2f*)&h[(long)(m0 + lid) * HID + k];
        v2f b;
        b.x = Wout[k * N_ACT + lid];
        b.y = Wout[(k + 1) * N_ACT + lid];
        acc = wmma_f32_k4(a, b, acc);
    }
    const float bv = bout[lid];
    #pragma unroll
    for (int r = 0; r < 8; ++r)
        out[(long)(m0 + 8 * half + r) * N_ACT + lid] = acc[r] + bv;
}

// ---------------------------------------------------------------------------
extern "C" void kernel_launch(void* const* d_in, const int* in_sizes, int n_in,
                              void* d_out, int out_size, void* d_ws, size_t ws_size,
                              hipStream_t stream) {
    const float* feat_gt    = (const float*)d_in[0];
    const float* feat_agent = (const float*)d_in[1];
    const float* h0         = (const float*)d_in[2];
    const int*   esrc       = (const int*)d_in[3];
    const int*   edst       = (const int*)d_in[4];
    const float* W_src      = (const float*)d_in[5];
    const float* b_src      = (const float*)d_in[6];
    const float* W_dst      = (const float*)d_in[7];
    const float* b_dst      = (const float*)d_in[8];
    const float* attn       = (const float*)d_in[9];
    const float* W_res      = (const float*)d_in[10];
    const float* b_res      = (const float*)d_in[11];
    const float* W_ih       = (const float*)d_in[12];
    const float* W_hh       = (const float*)d_in[13];
    const float* b_ih       = (const float*)d_in[14];
    const float* b_hh       = (const float*)d_in[15];
    const float* W_out      = (const float*)d_in[16];
    const float* b_out      = (const float*)d_in[17];

    float* out_logits = (float*)d_out;
    float* out_h      = out_logits + (size_t)N_DST * N_ACT;  // rst accum, then h

    float*    ws    = (float*)d_ws;
    float*    fs    = ws;                                    // N_SRC*HID
    float*    fd    = fs + (size_t)N_SRC * HID;              // N_DST*HID
    float*    resid = fd + (size_t)N_DST * HID;              // N_DST*HID
    float*    score = resid + (size_t)N_DST * HID;           // E*HEADS
    unsigned* smax  = (unsigned*)(score + (size_t)E_EDGES * HEADS);  // N_DST*HEADS
    float*    den   = (float*)(smax + (size_t)N_DST * HEADS);        // N_DST*HEADS

    proj_gemm<<<(N_SRC + 31) / 32, 256, 0, stream>>>(feat_gt,    W_src, b_src, fs, N_SRC);
    proj_gemm<<<(N_DST + 31) / 32, 256, 0, stream>>>(feat_agent, W_dst, b_dst, fd, N_DST);
    proj_gemm<<<(N_DST + 31) / 32, 256, 0, stream>>>(feat_agent, W_res, b_res, resid, N_DST);

    init_kernel<<<(N_DST * HID + 255) / 256, 256, 0, stream>>>(out_h, smax, den);

    edge_score<<<(E_EDGES + 7) / 8, 256, 0, stream>>>(fs, fd, attn, esrc, edst,
                                                      score, smax);
    edge_exp<<<(int)(((long)E_EDGES * HEADS + 255) / 256), 256, 0, stream>>>(
        score, smax, den, edst);
    edge_msg<<<(E_EDGES + 7) / 8, 256, 0, stream>>>(fs, score, den, esrc, edst,
                                                    out_h);

    gru_kernel<<<(N_DST + 31) / 32, 256, 0, stream>>>(resid, h0, W_ih, W_hh,
                                                      b_ih, b_hh, out_h);
    logits_kernel<<<(N_DST / 16 + 7) / 8, 256, 0, stream>>>(out_h, W_out, b_out,
                                                            out_logits);
}